// MedicalGNN_90606630076993
// MI455X (gfx1250) — compile-verified
//
#include <hip/hip_runtime.h>
#include <hip/hip_bf16.h>

typedef __attribute__((ext_vector_type(2))) float v2f;
typedef __attribute__((ext_vector_type(8))) float v8f;
typedef __attribute__((ext_vector_type(4))) int   v4i;

#define DIM 128

#if __has_builtin(__builtin_amdgcn_global_load_async_to_lds_b128) && \
    __has_builtin(__builtin_amdgcn_s_wait_asynccnt)
#define ASYNC_LDS 1
#else
#define ASYNC_LDS 0
#endif

// ---------------- degree / normalization ----------------
__global__ void k_fill1(float* __restrict__ deg, int N) {
  int i = blockIdx.x * blockDim.x + threadIdx.x;
  if (i < N) deg[i] = 1.0f;                 // self-loop contributes 1
}

__global__ void k_count(const int* __restrict__ dst, float* __restrict__ deg, int E) {
  int e = blockIdx.x * blockDim.x + threadIdx.x;
  if (e < E) atomicAdd(&deg[dst[e]], 1.0f);
}

__global__ void k_rsqrt(float* __restrict__ d, int N) {
  int i = blockIdx.x * blockDim.x + threadIdx.x;
  if (i < N) d[i] = rsqrtf(d[i]);           // deg >= 1 always
}

// ---------------- WMMA f32 GEMM:  out[N x DO] = act(A[x[i]] if GATHER) @ W[128 x DO] (+bias) ----
// One wave computes one 16x16 f32 tile via V_WMMA_F32_16X16X4_F32, K swept in steps of 4.
// A-frag layout (ISA 16x4 f32): lanes 0-15 -> K=0,1 ; lanes 16-31 -> K=2,3 ; M = lane&15.
// B-frag: lanes 0-15 -> K=0,1 rows ; lanes 16-31 -> K=2,3 rows ; N = lane&15.
// C/D: 8 VGPRs, N = lane&15, M = r (lanes 0-15) / r+8 (lanes 16-31).
// W panel staged into LDS via GLOBAL_LOAD_ASYNC_TO_LDS_B128 (ASYNCcnt-tracked).
template <int DO, bool RELU, bool BIAS, bool GATHER, bool TAIL>
__global__ __launch_bounds__(256) void k_gemm(const float* __restrict__ A,
                                              const float* __restrict__ W,
                                              const float* __restrict__ bias,
                                              const int*  __restrict__ xidx,
                                              float* __restrict__ out, int N) {
  __shared__ float Wl[DIM * DO];
  constexpr int CT = DO / 16;     // column tiles per block
  constexpr int RT = 8 / CT;      // row tiles per block (8 waves/block)

#if ASYNC_LDS
  // each thread moves 16B chunks straight into LDS, no VGPR bounce
#pragma unroll
  for (int base = threadIdx.x * 4; base < DIM * DO; base += 256 * 4) {
    __builtin_amdgcn_global_load_async_to_lds_b128(
        (__attribute__((address_space(1))) v4i*)(W + base),
        (__attribute__((address_space(3))) v4i*)(&Wl[base]), 0, 0);
  }
  __builtin_amdgcn_s_wait_asynccnt(0);
  __syncthreads();
#else
  for (int t = threadIdx.x; t < DIM * DO; t += 256) Wl[t] = W[t];
  __syncthreads();
#endif

  const int wave = threadIdx.x >> 5;
  const int lane = threadIdx.x & 31;
  const int rowTile = wave / CT;
  const int colTile = wave % CT;
  const int rowbase = blockIdx.x * (16 * RT) + rowTile * 16;
  const int colbase = colTile * 16;
  const int m  = lane & 15;       // matrix row within tile (A) / column (B,C,D)
  const int hi = lane >> 4;       // half-wave selects K pair 0,1 vs 2,3

  int row = rowbase + m;
  if (row >= N) row = N - 1;      // clamp: EXEC must stay all-ones for WMMA
  if (GATHER) row = xidx[row];
  const float* Arow = A + (size_t)row * DIM;
  __builtin_prefetch(Arow, 0, 0);

  const float bv = BIAS ? bias[colbase + m] : 0.0f;
  v8f acc;
#pragma unroll
  for (int r = 0; r < 8; ++r) acc[r] = bv;

#pragma unroll 4
  for (int k = 0; k < DIM; k += 4) {
    float a0 = Arow[k + 2 * hi];
    float a1 = Arow[k + 2 * hi + 1];
    if (RELU) { a0 = fmaxf(a0, 0.0f); a1 = fmaxf(a1, 0.0f); }
    v2f a; a[0] = a0; a[1] = a1;
    v2f b;
    b[0] = Wl[(k + 2 * hi) * DO + colbase + m];
    b[1] = Wl[(k + 2 * hi + 1) * DO + colbase + m];
    acc = __builtin_amdgcn_wmma_f32_16x16x4_f32(false, a, false, b,
                                                (short)0, acc, false, false);
  }

  const int m0 = hi ? 8 : 0;
#pragma unroll
  for (int r = 0; r < 8; ++r) {
    int orow = rowbase + m0 + r;
    if (!TAIL || orow < N) out[(size_t)orow * DO + colbase + m] = acc[r];
  }
}

// helper: pick the guard-free epilogue when the grid tiles N exactly
template <int DO, bool RELU, bool BIAS, bool GATHER>
static inline void launch_gemm(dim3 grid, const float* A, const float* W,
                               const float* bias, const int* xidx, float* out,
                               int N, hipStream_t stream) {
  constexpr int ROWS = (DO == DIM) ? 16 : 128;  // rows per block
  if (N % ROWS == 0)
    k_gemm<DO, RELU, BIAS, GATHER, false><<<grid, 256, 0, stream>>>(A, W, bias, xidx, out, N);
  else
    k_gemm<DO, RELU, BIAS, GATHER, true><<<grid, 256, 0, stream>>>(A, W, bias, xidx, out, N);
}

// ---------------- self-loop message + bias (no atomics needed) ----------------
// out[i,:] = b[:] + tmp[i,:] * dinv[i]^2
__global__ __launch_bounds__(DIM) void k_selfbias(const float* __restrict__ tmp,
                                                  const float* __restrict__ dinv,
                                                  const float* __restrict__ b,
                                                  float* __restrict__ out, int N) {
  int i = blockIdx.x;
  int j = threadIdx.x;
  float di = dinv[i];
  out[(size_t)i * DIM + j] = b[j] + tmp[(size_t)i * DIM + j] * di * di;
}

// ---------------- edge scatter-add: out[dst] += tmp[src] * dinv[src]*dinv[dst] ---------
// One wave per edge; each lane handles 4 contiguous columns (float4 load + 4 f32 atomics).
__global__ __launch_bounds__(256) void k_edge(const float* __restrict__ hin,
                                              const int* __restrict__ src,
                                              const int* __restrict__ dst,
                                              const float* __restrict__ dinv,
                                              float* __restrict__ out, int E) {
  int gw   = (blockIdx.x * 256 + threadIdx.x) >> 5;
  int lane = threadIdx.x & 31;
  if (gw >= E) return;
  int s = src[gw], d = dst[gw];
  float w = dinv[s] * dinv[d];
  const float4* hs = (const float4*)(hin + (size_t)s * DIM);
  float*        od = out + (size_t)d * DIM;
  float4 v = hs[lane];
  atomicAdd(&od[lane * 4 + 0], v.x * w);
  atomicAdd(&od[lane * 4 + 1], v.y * w);
  atomicAdd(&od[lane * 4 + 2], v.z * w);
  atomicAdd(&od[lane * 4 + 3], v.w * w);
}

// ---------------- host orchestration ----------------
extern "C" void kernel_launch(void* const* d_in, const int* in_sizes, int n_in,
                              void* d_out, int out_size, void* d_ws, size_t ws_size,
                              hipStream_t stream) {
  const int*   x    = (const int*)d_in[0];
  const int*   ei   = (const int*)d_in[1];
  const float* emb  = (const float*)d_in[2];
  const float* W0   = (const float*)d_in[3];
  const float* b0   = (const float*)d_in[4];
  const float* W1   = (const float*)d_in[5];
  const float* b1   = (const float*)d_in[6];
  const float* Wc   = (const float*)d_in[7];
  const float* bc   = (const float*)d_in[8];

  const int N = in_sizes[0];       // 50000
  const int E = in_sizes[1] / 2;   // 800000
  const int* src = ei;
  const int* dst = ei + E;

  // workspace layout (floats): dinv | tmp (N*D) | hbuf (N*D)
  float* ws   = (float*)d_ws;
  float* dinv = ws;
  float* tmp  = ws + (((size_t)N + 63) & ~(size_t)63);
  float* hbuf = tmp + (size_t)N * DIM;
  float* outf = (float*)d_out;

  const int T = 256;
  dim3 blkN((N + T - 1) / T), blkE((E + T - 1) / T);
  dim3 gemm128Grid((N + 15) / 16);           // exact for N=50000: 3125 blocks
  dim3 gemm16Grid((N + 127) / 128);
  dim3 edgeGrid((E + 7) / 8);                // one wave per edge, 8 waves/block

  // 1) symmetric degree normalization
  k_fill1<<<blkN, T, 0, stream>>>(dinv, N);
  k_count<<<blkE, T, 0, stream>>>(dst, dinv, E);
  k_rsqrt<<<blkN, T, 0, stream>>>(dinv, N);

  // 2) layer 0:  tmp = emb[x] @ W0 ; h1 = b0 + tmp*dinv^2 ; scatter edges
  launch_gemm<DIM, false, false, true>(gemm128Grid, emb, W0, nullptr, x, tmp, N, stream);
  k_selfbias<<<dim3(N), DIM, 0, stream>>>(tmp, dinv, b0, hbuf, N);
  k_edge<<<edgeGrid, T, 0, stream>>>(tmp, src, dst, dinv, hbuf, E);

  // 3) layer 1:  tmp = relu(h1) @ W1 ; h2 = b1 + tmp*dinv^2 ; scatter edges
  launch_gemm<DIM, true, false, false>(gemm128Grid, hbuf, W1, nullptr, nullptr, tmp, N, stream);
  k_selfbias<<<dim3(N), DIM, 0, stream>>>(tmp, dinv, b1, hbuf, N);
  k_edge<<<edgeGrid, T, 0, stream>>>(tmp, src, dst, dinv, hbuf, E);

  // 4) classifier:  out = relu(h2) @ Wc + bc   [N x 16]
  launch_gemm<16, true, true, false>(gemm16Grid, hbuf, Wc, bc, nullptr, outf, N, stream);
}